// Render_FLAME_79465484910658
// MI455X (gfx1250) — compile-verified
//
#include <hip/hip_runtime.h>
#include <hip/hip_bf16.h>
#include <stdint.h>

#define FOCALF 1015.0f
#define IMG_H 500
#define IMG_W 500
#define NVERT 5023
#define NFACE 9976
#define NBATCH 2
#define EPSF 1e-6f
#define NGROUP 2494          // NFACE / 4
#define NTILE 15625          // (500/4)*(500/4) 4x4 pixel tiles
#define TPR 125              // tiles per row
#define WPB 16               // waves per block (512 threads)
#define CHUNK 64             // face-groups staged in LDS per phase

typedef float v2f __attribute__((ext_vector_type(2)));
typedef float v8f __attribute__((ext_vector_type(8)));

__device__ __forceinline__ unsigned ballot32(bool p) {
#if __has_builtin(__builtin_amdgcn_ballot_w32)
    return __builtin_amdgcn_ballot_w32(p);
#else
    return (unsigned)__ballot(p);
#endif
}

__global__ void zero_normals_k(float* nrm, int n) {
    int i = blockIdx.x * blockDim.x + threadIdx.x;
    if (i < n) nrm[i] = 0.f;
}

__device__ __forceinline__ void cross3(const float a[3], const float b[3], float o[3]) {
    o[0] = a[1] * b[2] - a[2] * b[1];
    o[1] = a[2] * b[0] - a[0] * b[2];
    o[2] = a[0] * b[1] - a[1] * b[0];
}

__global__ void accum_normals_k(const float* __restrict__ verts,
                                const int* __restrict__ faces,
                                float* __restrict__ nrm) {
    int idx = blockIdx.x * blockDim.x + threadIdx.x;
    if (idx >= NBATCH * NFACE) return;
    int b = idx / NFACE, f = idx % NFACE;
    int i0 = faces[f * 3 + 0], i1 = faces[f * 3 + 1], i2 = faces[f * 3 + 2];
    const float* vb = verts + (size_t)b * NVERT * 3;
    float v0[3] = { vb[i0*3], vb[i0*3+1], vb[i0*3+2] };
    float v1[3] = { vb[i1*3], vb[i1*3+1], vb[i1*3+2] };
    float v2[3] = { vb[i2*3], vb[i2*3+1], vb[i2*3+2] };
    float d21[3] = { v2[0]-v1[0], v2[1]-v1[1], v2[2]-v1[2] };
    float d01[3] = { v0[0]-v1[0], v0[1]-v1[1], v0[2]-v1[2] };
    float d02[3] = { v0[0]-v2[0], v0[1]-v2[1], v0[2]-v2[2] };
    float d12[3] = { v1[0]-v2[0], v1[1]-v2[1], v1[2]-v2[2] };
    float d10[3] = { v1[0]-v0[0], v1[1]-v0[1], v1[2]-v0[2] };
    float d20[3] = { v2[0]-v0[0], v2[1]-v0[1], v2[2]-v0[2] };
    float c1[3], c2[3], c0[3];
    cross3(d21, d01, c1);   // -> vertex i1
    cross3(d02, d12, c2);   // -> vertex i2
    cross3(d10, d20, c0);   // -> vertex i0
    float* nb = nrm + (size_t)b * NVERT * 3;
    for (int c = 0; c < 3; ++c) {
        atomicAdd(&nb[i1*3 + c], c1[c]);
        atomicAdd(&nb[i2*3 + c], c2[c]);
        atomicAdd(&nb[i0*3 + c], c0[c]);
    }
}

__global__ void normalize_normals_k(float* nrm) {
    int i = blockIdx.x * blockDim.x + threadIdx.x;
    if (i >= NBATCH * NVERT) return;
    float* p = nrm + (size_t)i * 3;
    float l = sqrtf(p[0]*p[0] + p[1]*p[1] + p[2]*p[2]);
    float inv = 1.f / fmaxf(l, EPSF);
    p[0] *= inv; p[1] *= inv; p[2] *= inv;
}

// Per-face affine coefficients for (w0,w1,w2,z) packed in the A-operand
// (16x4 f32) lane layout of V_WMMA_F32_16X16X4_F32.
__global__ void face_setup_k(const float* __restrict__ verts,
                             const int* __restrict__ faces,
                             float* __restrict__ coef) {
    int idx = blockIdx.x * blockDim.x + threadIdx.x;
    if (idx >= NBATCH * NFACE) return;
    int b = idx / NFACE, f = idx % NFACE;
    int i0 = faces[f*3+0], i1 = faces[f*3+1], i2 = faces[f*3+2];
    const float* vb = verts + (size_t)b * NVERT * 3;
    float v0[3] = { vb[i0*3], vb[i0*3+1], vb[i0*3+2] };
    float v1[3] = { vb[i1*3], vb[i1*3+1], vb[i1*3+2] };
    float v2[3] = { vb[i2*3], vb[i2*3+1], vb[i2*3+2] };
    float zA = -v0[2], zB = -v1[2], zC = -v2[2];
    float ax = 0.5f*IMG_W - FOCALF * (-v0[0]) / fmaxf(zA, EPSF);
    float ay = 0.5f*IMG_H - FOCALF * ( v0[1]) / fmaxf(zA, EPSF);
    float bx = 0.5f*IMG_W - FOCALF * (-v1[0]) / fmaxf(zB, EPSF);
    float by = 0.5f*IMG_H - FOCALF * ( v1[1]) / fmaxf(zB, EPSF);
    float cx = 0.5f*IMG_W - FOCALF * (-v2[0]) / fmaxf(zC, EPSF);
    float cy = 0.5f*IMG_H - FOCALF * ( v2[1]) / fmaxf(zC, EPSF);
    float denom = (by - cy) * (ax - cx) + (cx - bx) * (ay - cy);
    bool valid = (fabsf(denom) >= EPSF) && (zA > EPSF) && (zB > EPSF) && (zC > EPSF);
    denom = valid ? denom : 1.f;
    float a0 = (by - cy) / denom, b0 = (cx - bx) / denom;
    float c0 = -(a0 * cx + b0 * cy);
    float a1 = (cy - ay) / denom, b1 = (ax - cx) / denom;
    float c1 = -(a1 * cx + b1 * cy);
    float a2 = -(a0 + a1), b2 = -(b0 + b1), c2 = 1.f - c0 - c1;
    float az = a0*zA + a1*zB + a2*zC;
    float bz = b0*zA + b1*zB + b2*zC;
    float cz = c0*zA + c1*zB + c2*zC;
    if (!valid) { // w0 == -1 everywhere -> never inside
        a0 = b0 = 0.f; c0 = -1.f;
        a1 = b1 = c1 = 0.f; a2 = b2 = c2 = 0.f; az = bz = cz = 0.f;
    }
    int g = f >> 2, m = (f & 3) * 4;
    float* base = coef + ((size_t)b * NGROUP + g) * 64;
    float wx[4] = { a0, a1, a2, az };
    float wc[4] = { c0, c1, c2, cz };
    float wy[4] = { b0, b1, b2, bz };
    for (int v = 0; v < 4; ++v) {
        base[m + v]      = wx[v];
        base[16 + m + v] = wc[v];
        base[32 + m + v] = wy[v];
        base[48 + m + v] = 0.f;
    }
}

// Screen-space bbox (union over the 4 faces of a group), padded by 0.5 px.
// Empty groups get an inverted bbox that never intersects.
__global__ void group_bbox_k(const float* __restrict__ verts,
                             const int* __restrict__ faces,
                             float* __restrict__ gbox) {
    int idx = blockIdx.x * blockDim.x + threadIdx.x;
    if (idx >= NBATCH * NGROUP) return;
    int b = idx / NGROUP, g = idx % NGROUP;
    const float* vb = verts + (size_t)b * NVERT * 3;
    float xmn = 3e38f, ymn = 3e38f, xmx = -3e38f, ymx = -3e38f;
    for (int v = 0; v < 4; ++v) {
        int f = g * 4 + v;
        int ii[3] = { faces[f*3+0], faces[f*3+1], faces[f*3+2] };
        float px[3], py[3], pz[3];
        for (int k = 0; k < 3; ++k) {
            const float* vv = vb + (size_t)ii[k] * 3;
            float z = -vv[2];
            pz[k] = z;
            float zs = fmaxf(z, EPSF);
            px[k] = 0.5f*IMG_W - FOCALF * (-vv[0]) / zs;
            py[k] = 0.5f*IMG_H - FOCALF * ( vv[1]) / zs;
        }
        float denom = (py[1]-py[2])*(px[0]-px[2]) + (px[2]-px[1])*(py[0]-py[2]);
        bool valid = (fabsf(denom) >= EPSF) &&
                     (pz[0] > EPSF) && (pz[1] > EPSF) && (pz[2] > EPSF);
        if (valid) {
            for (int k = 0; k < 3; ++k) {
                xmn = fminf(xmn, px[k]); xmx = fmaxf(xmx, px[k]);
                ymn = fminf(ymn, py[k]); ymx = fmaxf(ymx, py[k]);
            }
        }
    }
    float* o = gbox + (size_t)idx * 4;
    o[0] = xmn - 0.5f; o[1] = ymn - 0.5f; o[2] = xmx + 0.5f; o[3] = ymx + 0.5f;
}

__global__ __launch_bounds__(512)
void raster_shade_k(const float* __restrict__ verts,
                    const int* __restrict__ faces,
                    const float* __restrict__ nrm,
                    const float* __restrict__ coef,
                    const float* __restrict__ gbox,
                    float* __restrict__ out) {
    __shared__ float sCoef[CHUNK * 64];   // 16 KB
    __shared__ float sBox[CHUNK * 4];     // 1 KB
    int b = blockIdx.y;
    int lane = threadIdx.x & 31;
    int wave = threadIdx.x >> 5;
    int tile = blockIdx.x * WPB + wave;
    bool wvalid = tile < NTILE;
    int tx = tile % TPR, ty = tile / TPR;
    int x = tx * 4 + (lane & 3);
    int y = ty * 4 + ((lane >> 2) & 3);
    float xp = (float)x + 0.5f, yp = (float)y + 0.5f;
    float tx0 = (float)(tx * 4) + 0.5f, tx1 = tx0 + 3.f;
    float ty0 = (float)(ty * 4) + 0.5f, ty1 = ty0 + 3.f;

    // B operand (4x16): rows = (xp, yp, 1, 0); column n = pixel n of the tile
    v2f bv;
    bv[0] = (lane < 16) ? xp : 1.f;   // K=0 row (lanes 0-15), K=2 row (lanes 16-31)
    bv[1] = (lane < 16) ? yp : 0.f;   // K=1 row,               K=3 row

    const float* coefB = coef + (size_t)b * NGROUP * 64;
    const float* gboxB = gbox + (size_t)b * NGROUP * 4;
    unsigned long long best = ~0ULL;
    unsigned faceadd = (lane >= 16) ? 2u : 0u;

    for (int cstart = 0; cstart < NGROUP; cstart += CHUNK) {
        int count = min(CHUNK, NGROUP - cstart);
        __syncthreads();
        for (int i = threadIdx.x; i < count * 64; i += 512)
            sCoef[i] = coefB[(size_t)cstart * 64 + i];
        for (int i = threadIdx.x; i < count * 4; i += 512)
            sBox[i] = gboxB[(size_t)cstart * 4 + i];
        __syncthreads();
        if (!wvalid) continue;

        // Build one 64-bit survivor mask for the whole chunk (2 ballots).
        unsigned long long m;
        {
            int go = lane;
            bool p = false;
            if (go < count) {
                float4 bb = *((const float4*)&sBox[go * 4]); // (xmin,ymin,xmax,ymax)
                p = (bb.x <= tx1) & (bb.z >= tx0) & (bb.y <= ty1) & (bb.w >= ty0);
            }
            m = (unsigned long long)ballot32(p);
            go = 32 + lane;
            p = false;
            if (go < count) {
                float4 bb = *((const float4*)&sBox[go * 4]);
                p = (bb.x <= tx1) & (bb.z >= tx0) & (bb.y <= ty1) & (bb.w >= ty0);
            }
            m |= ((unsigned long long)ballot32(p)) << 32;
        }
        if (!m) continue;

        // Software-pipelined survivor loop: issue next LDS load before
        // consuming the current one, so only s_wait_dscnt <= 1 is needed.
        int gcur = __builtin_ctzll(m);
        m &= m - 1;
        v2f aCur;
        aCur[0] = sCoef[gcur * 64 + lane];
        aCur[1] = sCoef[gcur * 64 + 32 + lane];
        while (m) {
            int gnext = __builtin_ctzll(m);
            m &= m - 1;
            v2f aNext;
            aNext[0] = sCoef[gnext * 64 + lane];
            aNext[1] = sCoef[gnext * 64 + 32 + lane];

            v8f c = {};
            v8f d = __builtin_amdgcn_wmma_f32_16x16x4_f32(
                false, aCur, false, bv, (short)0, c, false, false);
            unsigned f0 = (unsigned)(cstart + gcur) * 4u + faceadd;
            bool in0 = (d[0] >= 0.f) & (d[1] >= 0.f) & (d[2] >= 0.f);
            unsigned hi0 = in0 ? __float_as_uint(fmaxf(d[3], 0.f)) : 0xFFFFFFFFu;
            unsigned lo0 = in0 ? f0 : 0xFFFFFFFFu;
            unsigned long long k0 = ((unsigned long long)hi0 << 32) | lo0;
            best = (k0 < best) ? k0 : best;
            bool in1 = (d[4] >= 0.f) & (d[5] >= 0.f) & (d[6] >= 0.f);
            unsigned hi1 = in1 ? __float_as_uint(fmaxf(d[7], 0.f)) : 0xFFFFFFFFu;
            unsigned lo1 = in1 ? (f0 + 1u) : 0xFFFFFFFFu;
            unsigned long long k1 = ((unsigned long long)hi1 << 32) | lo1;
            best = (k1 < best) ? k1 : best;

            gcur = gnext;
            aCur = aNext;
        }
        {   // drain the pipeline: process the last survivor
            v8f c = {};
            v8f d = __builtin_amdgcn_wmma_f32_16x16x4_f32(
                false, aCur, false, bv, (short)0, c, false, false);
            unsigned f0 = (unsigned)(cstart + gcur) * 4u + faceadd;
            bool in0 = (d[0] >= 0.f) & (d[1] >= 0.f) & (d[2] >= 0.f);
            unsigned hi0 = in0 ? __float_as_uint(fmaxf(d[3], 0.f)) : 0xFFFFFFFFu;
            unsigned lo0 = in0 ? f0 : 0xFFFFFFFFu;
            unsigned long long k0 = ((unsigned long long)hi0 << 32) | lo0;
            best = (k0 < best) ? k0 : best;
            bool in1 = (d[4] >= 0.f) & (d[5] >= 0.f) & (d[6] >= 0.f);
            unsigned hi1 = in1 ? __float_as_uint(fmaxf(d[7], 0.f)) : 0xFFFFFFFFu;
            unsigned lo1 = in1 ? (f0 + 1u) : 0xFFFFFFFFu;
            unsigned long long k1 = ((unsigned long long)hi1 << 32) | lo1;
            best = (k1 < best) ? k1 : best;
        }
    }
    if (!wvalid) return;
    unsigned long long other = __shfl_xor(best, 16, 32);
    best = (other < best) ? other : best;
    if (lane >= 16) return;

    bool hit = best != ~0ULL;
    float shade = 1.f, alpha = 0.f;
    if (hit) {
        unsigned fi = (unsigned)(best & 0xffffffffu);
        int g = fi >> 2, m = (fi & 3) * 4;
        const float* cb = coefB + (size_t)g * 64;
        float w0 = cb[m + 0] * xp + cb[32 + m + 0] * yp + cb[16 + m + 0];
        float w1 = cb[m + 1] * xp + cb[32 + m + 1] * yp + cb[16 + m + 1];
        float w2 = 1.f - w0 - w1;
        int i0 = faces[fi*3+0], i1 = faces[fi*3+1], i2 = faces[fi*3+2];
        const float* vb = verts + (size_t)b * NVERT * 3;
        const float* nb = nrm + (size_t)b * NVERT * 3;
        float P[3], N[3];
        for (int ch = 0; ch < 3; ++ch) {
            P[ch] = w0 * vb[i0*3+ch] + w1 * vb[i1*3+ch] + w2 * vb[i2*3+ch];
            N[ch] = w0 * nb[i0*3+ch] + w1 * nb[i1*3+ch] + w2 * nb[i2*3+ch];
        }
        float nl = sqrtf(N[0]*N[0] + N[1]*N[1] + N[2]*N[2]);
        float ninv = 1.f / fmaxf(nl, EPSF);
        N[0] *= ninv; N[1] *= ninv; N[2] *= ninv;
        float pl = sqrtf(P[0]*P[0] + P[1]*P[1] + P[2]*P[2]);
        float pinv = -1.f / fmaxf(pl, EPSF);
        float L[3] = { P[0]*pinv, P[1]*pinv, P[2]*pinv };
        float ln = N[0]*L[0] + N[1]*L[1] + N[2]*L[2];
        float ndl = fmaxf(ln, 0.f);
        float R[3] = { 2.f*ln*N[0] - L[0], 2.f*ln*N[1] - L[1], 2.f*ln*N[2] - L[2] };
        float vdr = fmaxf(L[0]*R[0] + L[1]*R[1] + L[2]*R[2], 0.f);
        float t = vdr;                      // vdr^64 via 6 squarings
        t *= t; t *= t; t *= t; t *= t; t *= t; t *= t;
        shade = 0.5f * (0.5f + 0.3f * ndl) + 0.2f * t;
        alpha = 1.f;
    }
    shade = fminf(fmaxf(shade, 0.f), 255.f);
    alpha = fminf(fmaxf(alpha, 0.f), 255.f);
    float4 o = make_float4(shade, shade, shade, alpha);
    *((float4*)(out + ((size_t)b * IMG_H * IMG_W + (size_t)y * IMG_W + x) * 4)) = o;
}

extern "C" void kernel_launch(void* const* d_in, const int* in_sizes, int n_in,
                              void* d_out, int out_size, void* d_ws, size_t ws_size,
                              hipStream_t stream) {
    const float* verts = (const float*)d_in[0];  // (B, NV, 3) f32
    const int*   faces = (const int*)d_in[1];    // (NF, 3) i32
    float* out = (float*)d_out;                  // (B, H, W, 4) f32

    float* nrm  = (float*)d_ws;                  // B*NV*3 = 30138 floats (pad to 30144)
    float* coef = nrm + 30144;                   // B*NGROUP*64 = 319232 floats
    float* gbox = coef + (size_t)NBATCH * NGROUP * 64; // B*NGROUP*4 = 19952 floats

    int nNrm = NBATCH * NVERT * 3;
    zero_normals_k<<<(nNrm + 255) / 256, 256, 0, stream>>>(nrm, nNrm);
    accum_normals_k<<<(NBATCH * NFACE + 255) / 256, 256, 0, stream>>>(verts, faces, nrm);
    normalize_normals_k<<<(NBATCH * NVERT + 255) / 256, 256, 0, stream>>>(nrm);
    face_setup_k<<<(NBATCH * NFACE + 255) / 256, 256, 0, stream>>>(verts, faces, coef);
    group_bbox_k<<<(NBATCH * NGROUP + 255) / 256, 256, 0, stream>>>(verts, faces, gbox);
    raster_shade_k<<<dim3((NTILE + WPB - 1) / WPB, NBATCH), 512, 0, stream>>>(
        verts, faces, nrm, coef, gbox, out);
}